// Gradient_28733331210651
// MI455X (gfx1250) — compile-verified
//
#include <hip/hip_runtime.h>
#include <hip/hip_bf16.h>

// y[b,c,h,w] = coef[c] * g[b,c,h,w]
// g: (32, 512, 64, 64) fp32  -> 64Mi elements, 256 MB in + 256 MB out.
// Pure HBM-streaming kernel: memory floor ~22us at 23.3 TB/s on MI455X.
//
// Strategy:
//  - one (b,c) plane of 4096 floats per 256-thread block (8 waves / wave32)
//  - 128-bit vector loads/stores (global_load_b128 / global_store_b128)
//  - coef[c] is block-uniform -> scalar load (s_load_b32), zero per-lane waste
//  - non-temporal hints: 512 MB stream > 192 MB L2, don't thrash the cache
//  - 4 independent b128 load/mul/store chains per thread to cover HBM latency

typedef float __attribute__((ext_vector_type(4))) v4f;

#define DIMC     512
#define PLANE_F  4096          // 64*64 floats per (b,c) plane
#define PLANE_F4 (PLANE_F / 4) // 1024 float4 per plane
#define BLOCK    256           // 8 wave32 per block; 4 float4 per thread

__global__ __launch_bounds__(BLOCK) void
chan_scale_kernel(const v4f* __restrict__ g,
                  const float* __restrict__ coef,
                  v4f* __restrict__ out) {
    const int plane = blockIdx.x;           // (b*512 + c), 16384 planes total
    const int c     = plane & (DIMC - 1);   // channel index, block-uniform
    const float s   = coef[c];              // uniform -> s_load_b32

    const long base = (long)plane * PLANE_F4;
    const int  t    = threadIdx.x;

#pragma unroll
    for (int i = 0; i < PLANE_F4 / BLOCK; ++i) {   // 4 iterations
        const long idx = base + t + (long)i * BLOCK;
        v4f v = __builtin_nontemporal_load(&g[idx]);  // global_load_b128, TH=NT
        v.x *= s;
        v.y *= s;
        v.z *= s;
        v.w *= s;
        __builtin_nontemporal_store(v, &out[idx]);    // global_store_b128, TH=NT
    }
}

extern "C" void kernel_launch(void* const* d_in, const int* in_sizes, int n_in,
                              void* d_out, int out_size, void* d_ws, size_t ws_size,
                              hipStream_t stream) {
    const v4f*   g    = (const v4f*)d_in[0];     // (32,512,64,64) fp32
    const float* coef = (const float*)d_in[1];   // (512,) fp32
    v4f*         out  = (v4f*)d_out;

    const int n_planes = in_sizes[0] / PLANE_F;  // 32*512 = 16384 blocks
    chan_scale_kernel<<<n_planes, BLOCK, 0, stream>>>(g, coef, out);
}